// MTopECE_31198642438677
// MI455X (gfx1250) — compile-verified
//
#include <hip/hip_runtime.h>
#include <math.h>

#define NBINS 15
#define COLS 1000
#define CHUNKS 250          // COLS / 4 (float4 chunks per row)
#define OPS_PER_ROW 8       // ceil(CHUNKS / 32) async b128 issues per row per wave
#define WPB 8               // waves per block (wave32)
#define BT (WPB * 32)       // 256 threads
#define MAXB 2048
#define PARTIALS_PER_BLOCK 45   // 15 counts + 15 sum_conf + 15 sum_acc

// ---------------- CDNA5 async global->LDS path (guarded) ----------------
#if defined(__HIP_DEVICE_COMPILE__) && defined(__gfx1250__) && \
    __has_builtin(__builtin_amdgcn_global_load_async_to_lds_b128) && \
    __has_builtin(__builtin_amdgcn_s_wait_asynccnt)
#define HAVE_ASYNC 1
#else
#define HAVE_ASYNC 0
#endif

#if HAVE_ASYNC
typedef int v4i __attribute__((ext_vector_type(4)));
typedef __attribute__((address_space(1))) v4i* gv4i_p;  // global v4i*
typedef __attribute__((address_space(3))) v4i* lv4i_p;  // LDS v4i*
__device__ __forceinline__ void async_cp16(const void* g, void* l) {
  // 16 bytes per active lane: global -> LDS, tracked by ASYNCcnt
  __builtin_amdgcn_global_load_async_to_lds_b128((gv4i_p)g, (lv4i_p)l, 0, 0);
}
#define WAIT_ASYNC(n)                                   \
  do {                                                  \
    __builtin_amdgcn_s_wait_asynccnt(n);                \
    __asm__ __volatile__("" ::: "memory");              \
  } while (0)
#endif

// Branch-free online softmax-denominator update: one v_exp_f32 per element.
// Maintains m = running max, s = sum(exp(x - m)), idx = first argmax index.
__device__ __forceinline__ void osm_update(float v, int gi, float& m, float& s,
                                           int& idx) {
  float mo = m;
  float mn = fmaxf(mo, v);
  float e  = __expf(fminf(v, mo) - mn);  // exp(smaller - larger) <= 1
  bool  g  = v > mo;
  s   = g ? fmaf(s, e, 1.0f) : (s + e);  // rescale-old+1  |  add-new-term
  idx = g ? gi : idx;
  m   = mn;
}

__global__ __launch_bounds__(BT) void ece_stage1_kernel(
    const float* __restrict__ logits, const int* __restrict__ labels,
    float* __restrict__ partials, int nrows, int totalWaves) {
  __shared__ float sred[WPB][48];
#if HAVE_ASYNC
  __shared__ __align__(16) float stage[WPB * 2 * 1024];  // 2 x 4KB per wave
#endif
  const int lane  = threadIdx.x & 31;
  const int wave  = threadIdx.x >> 5;
  const int gwave = blockIdx.x * WPB + wave;

  float cnt[NBINS], scf[NBINS], sac[NBINS];
#pragma unroll
  for (int b = 0; b < NBINS; ++b) { cnt[b] = 0.f; scf[b] = 0.f; sac[b] = 0.f; }

  const float fr = (float)nrows;

#if HAVE_ASYNC
  float* bufA = &stage[wave * 2048];
  float* bufB = bufA + 1024;
  if (gwave < nrows) {
    const float* g = logits + (size_t)gwave * COLS;
#pragma unroll
    for (int k = 0; k < OPS_PER_ROW; ++k) {
      int i = lane + k * 32;
      if (i < CHUNKS) async_cp16(g + 4 * i, bufA + 4 * i);
    }
  }
  int parity = 0;
#endif

  for (int row = gwave; row < nrows; row += totalWaves) {
#if HAVE_ASYNC
    int nxt = row + totalWaves;
    if (nxt < nrows) {
      const float* g  = logits + (size_t)nxt * COLS;
      float*       nb = parity ? bufA : bufB;
#pragma unroll
      for (int k = 0; k < OPS_PER_ROW; ++k) {
        int i = lane + k * 32;
        if (i < CHUNKS) async_cp16(g + 4 * i, nb + 4 * i);
      }
      WAIT_ASYNC(OPS_PER_ROW);  // in-order: current row's 8 DMAs are done
    } else {
      WAIT_ASYNC(0);
    }
    const float* src = parity ? bufB : bufA;
    parity ^= 1;
#else
    const float* src = logits + (size_t)row * COLS;
#endif

    float m = -__builtin_inff(), s = 0.f;
    int   idx = 0;
#pragma unroll
    for (int k = 0; k < OPS_PER_ROW; ++k) {
      int i = lane + k * 32;
      if (i < CHUNKS) {
        float4 v = *(const float4*)(src + 4 * i);
        int gi = 4 * i;
        osm_update(v.x, gi + 0, m, s, idx);
        osm_update(v.y, gi + 1, m, s, idx);
        osm_update(v.z, gi + 2, m, s, idx);
        osm_update(v.w, gi + 3, m, s, idx);
      }
    }

    // Cross-lane butterfly merge (wave32): combine (m, s), keep first argmax.
#pragma unroll
    for (int off = 16; off > 0; off >>= 1) {
      float m2 = __shfl_xor(m, off, 32);
      float s2 = __shfl_xor(s, off, 32);
      int   i2 = __shfl_xor(idx, off, 32);
      float M  = fmaxf(m, m2);
      s   = s * __expf(m - M) + s2 * __expf(m2 - M);
      idx = (m2 > m || (m2 == m && i2 < idx)) ? i2 : idx;
      m   = M;
    }

    float conf = 1.0f / s;                         // max(softmax(row))
    float acc  = (idx == labels[row]) ? 1.f : 0.f; // argmax == label

    // Faithful binning: bounds scaled by num_samples then rounded.
#pragma unroll
    for (int b = 0; b < NBINS; ++b) {
      float lo = roundf((float)b * fr * (1.0f / 15.0f));
      float up = roundf((float)(b + 1) * fr * (1.0f / 15.0f));
      bool  in = (conf > lo) && (conf <= up);
      cnt[b] += in ? 1.f : 0.f;
      scf[b] += in ? conf : 0.f;
      sac[b] += in ? acc : 0.f;
    }
  }

  // Deterministic block reduction: lane 0 of each wave spills, wave-parallel sum.
  if (lane == 0) {
#pragma unroll
    for (int b = 0; b < NBINS; ++b) {
      sred[wave][b]      = cnt[b];
      sred[wave][15 + b] = scf[b];
      sred[wave][30 + b] = sac[b];
    }
  }
  __syncthreads();
  if (threadIdx.x < PARTIALS_PER_BLOCK) {
    float a = 0.f;
#pragma unroll
    for (int w = 0; w < WPB; ++w) a += sred[w][threadIdx.x];
    partials[blockIdx.x * PARTIALS_PER_BLOCK + threadIdx.x] = a;
  }
}

__global__ __launch_bounds__(64) void ece_stage2_kernel(
    const float* __restrict__ partials, float* __restrict__ out, int nblocks,
    int nrows) {
  __shared__ double tot[PARTIALS_PER_BLOCK];
  int t = threadIdx.x;
  if (t < PARTIALS_PER_BLOCK) {
    double a = 0.0;
    for (int i = 0; i < nblocks; ++i)
      a += (double)partials[i * PARTIALS_PER_BLOCK + t];
    tot[t] = a;
  }
  __syncthreads();
  if (t == 0) {
    double ece = 0.0;
    for (int b = 0; b < NBINS; ++b) {
      double c  = tot[b];
      double sc = tot[15 + b];
      double sa = tot[30 + b];
      if (c > 0.0) {
        double prop = c / (double)nrows;
        ece += fabs(sc / c - sa / c) * prop;
      }
    }
    out[0] = (float)ece;
  }
}

extern "C" void kernel_launch(void* const* d_in, const int* in_sizes, int n_in,
                              void* d_out, int out_size, void* d_ws,
                              size_t ws_size, hipStream_t stream) {
  (void)n_in;
  (void)out_size;
  const float* logits = (const float*)d_in[0];
  const int*   labels = (const int*)d_in[1];
  float*       out    = (float*)d_out;
  float*       parts  = (float*)d_ws;

  int nrows = in_sizes[1];  // labels count == number of rows

  size_t cap = ws_size / (PARTIALS_PER_BLOCK * sizeof(float));
  int nblocks = (cap < (size_t)MAXB) ? (int)cap : MAXB;
  if (nblocks < 1) nblocks = 1;
  int totalWaves = nblocks * WPB;

  ece_stage1_kernel<<<nblocks, BT, 0, stream>>>(logits, labels, parts, nrows,
                                                totalWaves);
  ece_stage2_kernel<<<1, 64, 0, stream>>>(parts, out, nblocks, nrows);
}